// GraphConvolution_58059367907672
// MI455X (gfx1250) — compile-verified
//
#include <hip/hip_runtime.h>

typedef __attribute__((ext_vector_type(2))) float v2f;
typedef __attribute__((ext_vector_type(8))) float v8f;

#define D 128  // D_IN == D_OUT == 128 per reference

// ---------------------------------------------------------------------------
// Dense GEMM: P[N x 128] = X[N x 128] @ W[128 x 128] using fp32 WMMA.
// One wave computes one 16x16 tile of P; 8 waves per block cover all 8
// column tiles of a 16-row block. K accumulated in steps of 4 via
// V_WMMA_F32_16X16X4_F32 (32 WMMA ops per wave).
// ---------------------------------------------------------------------------
__global__ __launch_bounds__(256) void gcn_gemm_wmma(
    const float* __restrict__ X, const float* __restrict__ W,
    float* __restrict__ P, int nrows)
{
    const int wave = threadIdx.x >> 5;   // 0..7 -> column tile
    const int lane = threadIdx.x & 31;
    const int m0   = blockIdx.x << 4;    // row block (uniform)
    const int n0   = wave << 4;          // col block
    const int ml   = lane & 15;
    const int half = lane >> 4;          // 0: lanes 0-15, 1: lanes 16-31
    const int kb   = half << 1;          // A/B frag K sub-offset (0 or 2)

    // Clamp OOB rows for loads; WMMA needs EXEC all-ones. Each D row depends
    // only on its own A row, so clamped rows cannot pollute valid output.
    int mrow = m0 + ml;
    if (mrow >= nrows) mrow = nrows - 1;
    const float* __restrict__ xrow = X + (size_t)mrow * D;

    v8f acc = {};
    for (int k = 0; k < D; k += 4) {
        v2f a, b;
        // A 16x4 layout: VGPR0/1 hold K=kb, kb+1 for row m = lane%16
        a.x = xrow[k + kb];
        a.y = xrow[k + kb + 1];
        // B 4x16 layout (mirror): VGPR0/1 hold K=kb, kb+1 for col n = lane%16
        b.x = W[(size_t)(k + kb) * D + n0 + ml];
        b.y = W[(size_t)(k + kb + 1) * D + n0 + ml];
        acc = __builtin_amdgcn_wmma_f32_16x16x4_f32(
            /*neg_a=*/false, a, /*neg_b=*/false, b,
            /*c_mod=*/(short)0, acc, /*reuse_a=*/false, /*reuse_b=*/false);
    }

    // C/D layout: VGPR i -> row M = i + 8*half, col N = lane%16.
    // Uniform fast path: whole 16-row tile in range -> straight-line stores.
    float* __restrict__ pbase = P + (size_t)(m0 + (half << 3)) * D + n0 + ml;
    if (m0 + 16 <= nrows) {
        #pragma unroll
        for (int i = 0; i < 8; ++i)
            pbase[(size_t)i * D] = acc[i];
    } else {
        #pragma unroll
        for (int i = 0; i < 8; ++i) {
            int r = m0 + i + (half << 3);
            if (r < nrows)
                pbase[(size_t)i * D] = acc[i];
        }
    }
}

// ---------------------------------------------------------------------------
// Init out[i][:] = b[:] (bias broadcast; also clears the 0xAA poison).
// One float4 (16B) per thread -> global_store_b128.
// ---------------------------------------------------------------------------
__global__ void gcn_bias_init(float4* __restrict__ out,
                              const float4* __restrict__ b4, size_t total4)
{
    size_t t = (size_t)blockIdx.x * blockDim.x + threadIdx.x;
    if (t < total4) out[t] = b4[t & (D / 4 - 1)];
}

// ---------------------------------------------------------------------------
// COO SpMM scatter: out[rows[e]] += vals[e] * pre[cols[e]].
// One wave per edge; each lane handles one 16B chunk (float4) of the 512B
// row -> fully coalesced gather of one pre row, 4 hardware f32 atomics per
// lane into one out row. pre0+pre1+out (154MB) are L2-resident (192MB L2),
// so gathers/atomics run at L2 bandwidth.
// ---------------------------------------------------------------------------
__global__ void gcn_spmm_atomic(const int* __restrict__ rows,
                                const int* __restrict__ cols,
                                const float* __restrict__ vals,
                                const float* __restrict__ pre,
                                float* __restrict__ out, int nedges)
{
    size_t t = (size_t)blockIdx.x * blockDim.x + threadIdx.x;
    int e = (int)(t >> 5);
    if (e >= nedges) return;
    int q = (int)(t & 31);                 // float4 chunk index (0..31)

    int   r = rows[e];                     // wave-uniform
    int   c = cols[e];
    float v = vals[e];

    const float4* __restrict__ src = (const float4*)(pre + (size_t)c * D);
    float4 g = src[q];

    float* dst = out + (size_t)r * D + q * 4;
    unsafeAtomicAdd(dst + 0, v * g.x);     // global_atomic_add_f32, no return
    unsafeAtomicAdd(dst + 1, v * g.y);
    unsafeAtomicAdd(dst + 2, v * g.z);
    unsafeAtomicAdd(dst + 3, v * g.w);
}

// ---------------------------------------------------------------------------
// In-place ReLU, float4-wide (global_load_b128 / global_store_b128).
// ---------------------------------------------------------------------------
__global__ void gcn_relu(float4* __restrict__ out, size_t total4)
{
    size_t t = (size_t)blockIdx.x * blockDim.x + threadIdx.x;
    if (t < total4) {
        float4 v = out[t];
        v.x = v.x > 0.0f ? v.x : 0.0f;
        v.y = v.y > 0.0f ? v.y : 0.0f;
        v.z = v.z > 0.0f ? v.z : 0.0f;
        v.w = v.w > 0.0f ? v.w : 0.0f;
        out[t] = v;
    }
}

// ---------------------------------------------------------------------------
// inputs: x, rows0, cols0, vals0, rows1, cols1, vals1, W0, W1, b
// ---------------------------------------------------------------------------
extern "C" void kernel_launch(void* const* d_in, const int* in_sizes, int n_in,
                              void* d_out, int out_size, void* d_ws, size_t ws_size,
                              hipStream_t stream)
{
    const float* x     = (const float*)d_in[0];
    const int*   rows0 = (const int*)  d_in[1];
    const int*   cols0 = (const int*)  d_in[2];
    const float* vals0 = (const float*)d_in[3];
    const int*   rows1 = (const int*)  d_in[4];
    const int*   cols1 = (const int*)  d_in[5];
    const float* vals1 = (const float*)d_in[6];
    const float* W0    = (const float*)d_in[7];
    const float* W1    = (const float*)d_in[8];
    const float* b     = (const float*)d_in[9];

    const int N = in_sizes[0] / D;
    const int E = in_sizes[1];

    float* pre0 = (float*)d_ws;                 // N*128 f32 = 51.2 MB
    float* pre1 = pre0 + (size_t)N * D;         // N*128 f32 = 51.2 MB
    float* out  = (float*)d_out;

    const int    tb     = 256;
    const size_t total4 = (size_t)N * (D / 4);
    const unsigned eb   = (unsigned)((total4 + tb - 1) / tb);

    // 1) out = b (broadcast, 128-bit stores)
    gcn_bias_init<<<eb, tb, 0, stream>>>((float4*)out, (const float4*)b, total4);

    // 2) pre0 = X@W0, pre1 = X@W1 (WMMA fp32)
    const unsigned gblocks = (unsigned)((N + 15) / 16);
    gcn_gemm_wmma<<<gblocks, 256, 0, stream>>>(x, W0, pre0, N);
    gcn_gemm_wmma<<<gblocks, 256, 0, stream>>>(x, W1, pre1, N);

    // 3) out += A0 @ pre0 ; out += A1 @ pre1  (atomic scatter, 1 wave/edge)
    const size_t   sth     = (size_t)E * 32;
    const unsigned sblocks = (unsigned)((sth + tb - 1) / tb);
    gcn_spmm_atomic<<<sblocks, tb, 0, stream>>>(rows0, cols0, vals0, pre0, out, E);
    gcn_spmm_atomic<<<sblocks, tb, 0, stream>>>(rows1, cols1, vals1, pre1, out, E);

    // 4) out = relu(out) (128-bit load/store)
    gcn_relu<<<eb, tb, 0, stream>>>((float4*)out, total4);
}